// FCNN_63651415326860
// MI455X (gfx1250) — compile-verified
//
#include <hip/hip_runtime.h>

#define IMG_H 2048
#define IMG_W 2048
#define TILE_H 32
#define NBUF 8            // ring of 8 row buffers -> 64 KB LDS
#define NTHREADS 256

typedef float v4f __attribute__((ext_vector_type(4)));

// Async copy of 16 bytes from global memory into LDS (gfx1250 ASYNCcnt path).
// vdst = 32-bit LDS byte address, vaddr = 64-bit global address.
#define ASYNC_LD_B128(lds_addr, gptr)                                   \
  asm volatile("global_load_async_to_lds_b128 %0, %1, off"              \
               :: "v"(lds_addr), "v"(gptr) : "memory")

#define WAIT_ASYNC(n) asm volatile("s_wait_asynccnt " #n ::: "memory")

__device__ __forceinline__ void proc4(int c,
                                      const float* __restrict__ up,
                                      const float* __restrict__ ctr,
                                      const float* __restrict__ dn,
                                      float* __restrict__ orow,
                                      float a0, float a1, float a2,
                                      float w0, float w1, float w2, float w3) {
  v4f xc = *(const v4f*)(ctr + c);
  v4f xu = *(const v4f*)(up + c);
  v4f xd = *(const v4f*)(dn + c);
  int cl = (c - 1 < 0) ? 0 : c - 1;                 // replicate left edge
  int cr = (c + 4 > IMG_W - 1) ? IMG_W - 1 : c + 4; // replicate right edge
  float xl = ctr[cl];
  float xr = ctr[cr];

  float hx = xl   + xc.y;
  float hy = xc.x + xc.z;
  float hz = xc.y + xc.w;
  float hw = xc.z + xr;

  float vx = xu.x + xd.x;
  float vy = xu.y + xd.y;
  float vz = xu.z + xd.z;
  float vw = xu.w + xd.w;

  v4f o;
  // stencil = a0*center + a1*horiz + a2*vert ; poly = ((w3*d+w2)*d+w1)*d+w0, d = x
  {
    float p = fmaf(fmaf(fmaf(w3, xc.x, w2), xc.x, w1), xc.x, w0);
    o.x = fmaf(a0, xc.x, fmaf(a1, hx, fmaf(a2, vx, p)));
  }
  {
    float p = fmaf(fmaf(fmaf(w3, xc.y, w2), xc.y, w1), xc.y, w0);
    o.y = fmaf(a0, xc.y, fmaf(a1, hy, fmaf(a2, vy, p)));
  }
  {
    float p = fmaf(fmaf(fmaf(w3, xc.z, w2), xc.z, w1), xc.z, w0);
    o.z = fmaf(a0, xc.z, fmaf(a1, hz, fmaf(a2, vz, p)));
  }
  {
    float p = fmaf(fmaf(fmaf(w3, xc.w, w2), xc.w, w1), xc.w, w0);
    o.w = fmaf(a0, xc.w, fmaf(a1, hw, fmaf(a2, vw, p)));
  }
  // Output stream has zero reuse: bypass-rinse it (NT) so the 192MB L2 keeps
  // the input rows for halo reuse by vertically adjacent blocks.
  __builtin_nontemporal_store(o, (v4f*)(orow + c));
}

__global__ __launch_bounds__(NTHREADS) void FCNN_63651415326860_kernel(
    const float* __restrict__ x, const float* __restrict__ a,
    const float* __restrict__ w, float* __restrict__ out) {
  __shared__ float smem[NBUF * IMG_W]; // 64 KB ring of row buffers

  const int t  = threadIdx.x;
  const int r0 = blockIdx.x * TILE_H;           // first output row of strip
  const int b  = blockIdx.y;                    // image index

  const float* img  = x   + (size_t)b * IMG_H * IMG_W;
  float*       outp = out + (size_t)b * IMG_H * IMG_W;

  const float a0 = a[0], a1 = a[1], a2 = a[2];
  const float w0 = w[0], w1 = w[1], w2 = w[2], w3 = w[3];

  // Low 32 bits of a generic (flat) shared pointer are the LDS byte address.
  const unsigned lds_base = (unsigned)(uintptr_t)(&smem[0]);

  const int c1 = 4 * t;          // columns [0,1020]   (16B-aligned)
  const int c2 = 4 * t + 1024;   // columns [1024,2044]

  // Issue async load of logical row slot k (row r0+k-1, vertically clamped).
  auto issue_row = [&](int k) {
    int lr = r0 + k - 1;
    lr = lr < 0 ? 0 : (lr >= IMG_H ? IMG_H - 1 : lr);
    const float* g = img + (size_t)lr * IMG_W;
    unsigned lbuf = lds_base + (unsigned)((k & (NBUF - 1)) * IMG_W * 4);
    ASYNC_LD_B128(lbuf + (unsigned)(c1 * 4), g + c1);
    ASYNC_LD_B128(lbuf + (unsigned)(c2 * 4), g + c2);
  };

  auto compute_row = [&](int i) {
    const float* up  = &smem[((i)     & (NBUF - 1)) * IMG_W];
    const float* ctr = &smem[((i + 1) & (NBUF - 1)) * IMG_W];
    const float* dn  = &smem[((i + 2) & (NBUF - 1)) * IMG_W];
    float* orow = outp + (size_t)(r0 + i) * IMG_W;
    proc4(c1, up, ctr, dn, orow, a0, a1, a2, w0, w1, w2, w3);
    proc4(c2, up, ctr, dn, orow, a0, a1, a2, w0, w1, w2, w3);
  };

  // Prologue: rows r0-1, r0, r0+1 (slots 0,1,2) -> 6 outstanding per wave.
  issue_row(0);
  issue_row(1);
  issue_row(2);

  // Steady state: stay 2 rows ahead. Slot (i+3) overwrites a buffer 8 slots
  // stale; every wave has passed barrier i-1, so the slowest wave can at most
  // be computing row i-1 (reading slots i-1..i+1) -> reuse is race-free.
  #pragma unroll 1
  for (int i = 0; i < TILE_H - 1; ++i) {
    issue_row(i + 3);
    WAIT_ASYNC(2);   // retire everything up through slot i+2 (in-order)
    __syncthreads(); // make all waves' async row data visible
    compute_row(i);
  }

  // Epilogue: last row, drain all outstanding async traffic.
  WAIT_ASYNC(0);
  __syncthreads();
  compute_row(TILE_H - 1);
}

extern "C" void kernel_launch(void* const* d_in, const int* in_sizes, int n_in,
                              void* d_out, int out_size, void* d_ws, size_t ws_size,
                              hipStream_t stream) {
  const float* x = (const float*)d_in[0];
  const float* a = (const float*)d_in[1];
  const float* w = (const float*)d_in[2];
  float* out = (float*)d_out;

  const int batches = out_size / (IMG_H * IMG_W); // 16 for the reference shape
  dim3 grid(IMG_H / TILE_H, batches);
  FCNN_63651415326860_kernel<<<grid, NTHREADS, 0, stream>>>(x, a, w, out);
}